// EdgeConv_90185723281823
// MI455X (gfx1250) — compile-verified
//
#include <hip/hip_runtime.h>
#include <cstddef>

// EdgeConv for MI455X (gfx1250): f32 WMMA 16x16x4 everywhere.
//   B=4, N=4096, C_IN=128, C_HID=128, C_OUT=256, K=16
// Pipeline: norms / weight+BN packing -> (dist GEMM + top-16 via WMMA)
//           -> (edge MLP + max via WMMA, packed-weight b64 B-fragments)

#define BB 4
#define NN 4096
#define CIN 128
#define CHID 128
#define COUT 256
#define KNN 16
#define EPSV 1e-5f

typedef __attribute__((ext_vector_type(2))) float v2f;
typedef __attribute__((ext_vector_type(8))) float v8f;

__device__ __forceinline__ v8f wmma4(v2f a, v2f b, v8f c) {
  // D = A(16x4 f32) * B(4x16 f32) + C(16x16 f32)
  return __builtin_amdgcn_wmma_f32_16x16x4_f32(false, a, false, b, (short)0, c,
                                               false, false);
}

__device__ __forceinline__ v8f vzero8() {
  v8f z = {0.f, 0.f, 0.f, 0.f, 0.f, 0.f, 0.f, 0.f};
  return z;
}

// ---------------------------------------------------------------- kernel 1
// sq[p] = ||x[p]||^2  (p = b*N + n)
__global__ __launch_bounds__(256) void norms_kernel(const float* __restrict__ x,
                                                    float* __restrict__ sq) {
  const int p = blockIdx.x * 256 + threadIdx.x;
  const float4* r = (const float4*)(x + (size_t)p * CIN);
  float s = 0.f;
#pragma unroll
  for (int i = 0; i < CIN / 4; ++i) {
    float4 v = r[i];
    s += v.x * v.x + v.y * v.y + v.z * v.z + v.w * v.w;
  }
  sq[p] = s;
}

// ---------------------------------------------------------------- kernel 1b
// Pack weight row-pairs so a WMMA B-fragment is ONE b64 load:
//   W1p[pair*CHID + d] = { W1[2*pair][d], W1[2*pair+1][d] }   (128 pairs)
//   W2p[pair*COUT + e] = { W2[2*pair][e], W2[2*pair+1][e] }   (64 pairs)
// Both arrays have 16384 float2 elements.
__global__ __launch_bounds__(256) void pack_w_kernel(
    const float* __restrict__ W1, const float* __restrict__ W2,
    float* __restrict__ W1p, float* __restrict__ W2p) {
  const int i = blockIdx.x * 256 + threadIdx.x;  // [0, 16384)
  {
    const int pair = i >> 7, d = i & (CHID - 1);
    W1p[2 * i + 0] = W1[(size_t)(2 * pair + 0) * CHID + d];
    W1p[2 * i + 1] = W1[(size_t)(2 * pair + 1) * CHID + d];
  }
  {
    const int pair = i >> 8, e = i & (COUT - 1);
    W2p[2 * i + 0] = W2[(size_t)(2 * pair + 0) * COUT + e];
    W2p[2 * i + 1] = W2[(size_t)(2 * pair + 1) * COUT + e];
  }
}

// ---------------------------------------------------------------- kernel 1c
// Fold BN (+conv bias) into a per-channel affine: y*scale + shift, packed f2.
__global__ __launch_bounds__(384) void pack_bn_kernel(
    const float* __restrict__ b1, const float* __restrict__ g1,
    const float* __restrict__ be1, const float* __restrict__ mu1,
    const float* __restrict__ var1, const float* __restrict__ b2,
    const float* __restrict__ g2, const float* __restrict__ be2,
    const float* __restrict__ mu2, const float* __restrict__ var2,
    float* __restrict__ P1, float* __restrict__ P2) {
  const int t = threadIdx.x;
  if (t < CHID) {
    const float s = g1[t] * rsqrtf(var1[t] + EPSV);
    P1[2 * t + 0] = s;
    P1[2 * t + 1] = (b1[t] - mu1[t]) * s + be1[t];
  } else {
    const int e = t - CHID;
    const float s = g2[e] * rsqrtf(var2[e] + EPSV);
    P2[2 * e + 0] = s;
    P2[2 * e + 1] = (b2[e] - mu2[e]) * s + be2[e];
  }
}

// ---------------------------------------------------------------- kernel 2
// One wave per 16-row strip. G = Xr(16x128) @ Xc^T(128x16) via 32 wmma/tile.
// Selection key = sq[j] - 2*dot (row-constant sq_i cancels in ranking).
__global__ __launch_bounds__(32) void knn_kernel(const float* __restrict__ x,
                                                 const float* __restrict__ sq,
                                                 int* __restrict__ knn) {
  __shared__ float dt[16][17];  // padded transpose buffer
  const int lane = threadIdx.x;
  const int half = lane >> 4;
  const int j = lane & 15;
  const int b = blockIdx.x >> 8;           // 256 row tiles per batch
  const int n0 = (blockIdx.x & 255) << 4;  // first row of strip

  // Preload A fragments for the whole strip: 32 x v2f = 64 VGPRs, reused 256x.
  const float* arow = x + ((size_t)b * NN + n0 + j) * CIN;
  v2f af[32];
#pragma unroll
  for (int k = 0; k < 32; ++k) af[k] = *(const v2f*)(arow + 4 * k + 2 * half);

  float best[KNN];
  int bidx[KNN];
#pragma unroll
  for (int i = 0; i < KNN; ++i) {
    best[i] = 3.4e38f;
    bidx[i] = 0;
  }

#pragma unroll 1
  for (int m0 = 0; m0 < NN; m0 += 16) {
    const float* brow = x + ((size_t)b * NN + m0 + j) * CIN;
    v8f acc0 = vzero8(), acc1 = vzero8();  // two independent wmma chains
#pragma unroll
    for (int k = 0; k < 16; ++k) {
      v2f bf = *(const v2f*)(brow + 4 * k + 2 * half);
      acc0 = wmma4(af[k], bf, acc0);
    }
#pragma unroll
    for (int k = 16; k < 32; ++k) {
      v2f bf = *(const v2f*)(brow + 4 * k + 2 * half);
      acc1 = wmma4(af[k], bf, acc1);
    }
    const float sqj = sq[(size_t)b * NN + m0 + j];

    __syncthreads();  // protect dt WAR from previous iteration's readers
#pragma unroll
    for (int v = 0; v < 8; ++v)
      dt[v + 8 * half][j] = sqj - 2.0f * (acc0[v] + acc1[v]);
    __syncthreads();

    if (lane < 16) {  // lane r owns row r: sorted-insert top-16
#pragma unroll
      for (int jj = 0; jj < 16; ++jj) {
        float d = dt[lane][jj];
        if (d < best[KNN - 1]) {
          float cd = d;
          int ci = m0 + jj;
#pragma unroll
          for (int i = 0; i < KNN; ++i) {
            bool lt = cd < best[i];
            float tb = best[i];
            int ti = bidx[i];
            best[i] = lt ? cd : tb;
            bidx[i] = lt ? ci : ti;
            cd = lt ? tb : cd;
            ci = lt ? ti : ci;
          }
        }
      }
    }
  }

  if (lane < 16) {
    int* o = knn + ((size_t)b * NN + n0 + lane) * KNN;
#pragma unroll
    for (int i = 0; i < KNN; ++i) o[i] = bidx[i];
  }
}

// ---------------------------------------------------------------- kernel 3
// One wave per point. edge(16x256) @ W1 -> BN+ReLU -> H (LDS, pitch 130)
// H(16x128) @ W2 -> BN -> max over 16 rows -> out[b][n][256].
// B-fragments come from the packed weights: one b64 load per WMMA.
__global__ __launch_bounds__(128) void mlp_kernel(
    const float* __restrict__ x, const int* __restrict__ knn,
    const float* __restrict__ W1p, const float* __restrict__ W2p,
    const float* __restrict__ P1, const float* __restrict__ P2,
    float* __restrict__ out) {
  __shared__ float Hs[4 * 16 * 130];  // 4 waves x (16 rows x pitch 130)
  const int lane = threadIdx.x & 31;
  const int wave = threadIdx.x >> 5;
  const int half = lane >> 4;
  const int j = lane & 15;
  const int p = blockIdx.x * 4 + wave;  // point id in [0, B*N)
  const int b = p >> 12;                // N = 4096
  float* Hw = &Hs[wave * 16 * 130];

  const v2f* W1v = (const v2f*)W1p;
  const v2f* W2v = (const v2f*)W2p;

  const float* ctr = x + (size_t)p * CIN;              // edge channels 0..127
  const int nb = knn[(size_t)p * KNN + j];             // my A-row's neighbor
  const float* nbr = x + ((size_t)b * NN + nb) * CIN;  // edge channels 128..255

  // ---------- MLP1: 8 output tiles accumulated together ----------
  v8f acc[8];
#pragma unroll
  for (int t = 0; t < 8; ++t) acc[t] = vzero8();

#pragma unroll 1
  for (int k = 0; k < 32; ++k) {  // K = 0..127: center features (row-broadcast)
    const int c = 4 * k + 2 * half;
    const int pbase = (2 * k + half) * CHID;  // pair index base into W1p
    v2f a = *(const v2f*)(ctr + c);
#pragma unroll
    for (int t = 0; t < 8; ++t) {
      v2f bf = W1v[pbase + 16 * t + j];
      acc[t] = wmma4(a, bf, acc[t]);
    }
  }
#pragma unroll 1
  for (int k = 0; k < 32; ++k) {  // K = 128..255: gathered neighbor features
    const int c = 4 * k + 2 * half;
    const int pbase = (2 * k + half + 64) * CHID;
    v2f a = *(const v2f*)(nbr + c);
#pragma unroll
    for (int t = 0; t < 8; ++t) {
      v2f bf = W1v[pbase + 16 * t + j];
      acc[t] = wmma4(a, bf, acc[t]);
    }
  }

  // BN + ReLU fused in C layout (lane's channel d is constant), stage H in LDS
#pragma unroll
  for (int t = 0; t < 8; ++t) {
    const int d = 16 * t + j;
    const v2f pp = *(const v2f*)(P1 + 2 * d);  // {scale, shift}
#pragma unroll
    for (int v = 0; v < 8; ++v)
      Hw[(v + 8 * half) * 130 + d] = fmaxf(acc[t][v] * pp[0] + pp[1], 0.f);
  }
  __syncthreads();

  // ---------- MLP2: 16 output tiles in two passes of 8 ----------
#pragma unroll 1
  for (int pass = 0; pass < 2; ++pass) {
    v8f acc2[8];
#pragma unroll
    for (int tt = 0; tt < 8; ++tt) acc2[tt] = vzero8();

#pragma unroll 1
    for (int k = 0; k < 32; ++k) {
      const int c = 4 * k + 2 * half;
      const int pbase = (2 * k + half) * COUT;
      v2f a = *(const v2f*)(Hw + j * 130 + c);  // pitch 130 -> conflict-free
#pragma unroll
      for (int tt = 0; tt < 8; ++tt) {
        const int t = pass * 8 + tt;
        v2f bf = W2v[pbase + 16 * t + j];
        acc2[tt] = wmma4(a, bf, acc2[tt]);
      }
    }
#pragma unroll
    for (int tt = 0; tt < 8; ++tt) {
      const int t = pass * 8 + tt;
      const int e = 16 * t + j;
      const v2f pp = *(const v2f*)(P2 + 2 * e);  // {scale, shift}
      float mx = acc2[tt][0] * pp[0] + pp[1];
#pragma unroll
      for (int v = 1; v < 8; ++v) mx = fmaxf(mx, acc2[tt][v] * pp[0] + pp[1]);
      mx = fmaxf(mx, __shfl_xor(mx, 16, 32));  // rows 0-7 vs rows 8-15
      if (lane < 16) out[(size_t)p * COUT + e] = mx;
    }
  }
}

// ---------------------------------------------------------------- launch
extern "C" void kernel_launch(void* const* d_in, const int* in_sizes, int n_in,
                              void* d_out, int out_size, void* d_ws,
                              size_t ws_size, hipStream_t stream) {
  const float* x = (const float*)d_in[0];
  const float* W1 = (const float*)d_in[1];
  const float* b1 = (const float*)d_in[2];
  const float* g1 = (const float*)d_in[3];
  const float* be1 = (const float*)d_in[4];
  const float* mu1 = (const float*)d_in[5];
  const float* var1 = (const float*)d_in[6];
  const float* W2 = (const float*)d_in[7];
  const float* b2 = (const float*)d_in[8];
  const float* g2 = (const float*)d_in[9];
  const float* be2 = (const float*)d_in[10];
  const float* mu2 = (const float*)d_in[11];
  const float* var2 = (const float*)d_in[12];
  float* out = (float*)d_out;

  // Workspace layout (bytes):
  char* ws = (char*)d_ws;
  float* sq = (float*)ws;                                 //   64 KB
  int* knn = (int*)(ws + 65536);                          // 1024 KB
  float* W1p = (float*)(ws + 65536 + 1048576);            //  128 KB
  float* W2p = (float*)(ws + 65536 + 1048576 + 131072);   //  128 KB
  float* P1 = (float*)(ws + 65536 + 1048576 + 262144);    //    1 KB
  float* P2 = (float*)(ws + 65536 + 1048576 + 262144 + 1024);  // 2 KB

  norms_kernel<<<(BB * NN) / 256, 256, 0, stream>>>(x, sq);
  pack_w_kernel<<<16384 / 256, 256, 0, stream>>>(W1, W2, W1p, W2p);
  pack_bn_kernel<<<1, 384, 0, stream>>>(b1, g1, be1, mu1, var1, b2, g2, be2,
                                        mu2, var2, P1, P2);
  knn_kernel<<<BB * (NN / 16), 32, 0, stream>>>(x, sq, knn);
  mlp_kernel<<<(BB * NN) / 4, 128, 0, stream>>>(x, knn, W1p, W2p, P1, P2, out);
}